// ConvSkipLayer_52862457479531
// MI455X (gfx1250) — compile-verified
//
#include <hip/hip_runtime.h>
#include <hip/hip_bf16.h>

#define HH 8
#define CC 16
#define HC 128     // H*C
#define INCH 128
#define NEG_SLOPE 0.2f

typedef __attribute__((ext_vector_type(2))) float v2f;
typedef __attribute__((ext_vector_type(8))) float v8f;

// ---------------------------------------------------------------------------
// Kernel 1: fused dual GEMM (xh = x@lin_w, base = x@skip_w + skip_b + gat_bias)
// using f32 WMMA 16x16x4. One block = 16 nodes x 128 cols; wave w owns column
// tile [16w, 16w+16) == head w. Also produces a_src/a_dst head dots via
// in-register shuffle reduction over the D-fragment.
// ---------------------------------------------------------------------------
__global__ __launch_bounds__(256)
void gat_gemm_kernel(const float* __restrict__ x,
                     const float* __restrict__ lin_w,
                     const float* __restrict__ skip_w,
                     const float* __restrict__ att_src,
                     const float* __restrict__ att_dst,
                     const float* __restrict__ gat_bias,
                     const float* __restrict__ skip_b,
                     float* __restrict__ xh,
                     float* __restrict__ a_src,
                     float* __restrict__ a_dst,
                     float* __restrict__ base_out) {
  const int wave  = threadIdx.x >> 5;        // 0..7 -> head / 16-col tile
  const int lane  = threadIdx.x & 31;
  const int m16   = lane & 15;               // row (A) / col (B,C,D) within tile
  const int khalf = (lane >> 4) << 1;        // 0 or 2: K-half selector
  const long row0 = (long)blockIdx.x * 16;
  const int  ncol = wave * 16 + m16;         // global output column

  v8f acc_lin  = {};
  v8f acc_skip = {};

  const float* xrow = x + (row0 + m16) * INCH;

  #pragma unroll 4
  for (int k = 0; k < INCH; k += 4) {
    // A fragment: 16x4 f32; lane holds x[row=m16, k+khalf .. k+khalf+1]
    v2f a;
    a.x = xrow[k + khalf];
    a.y = xrow[k + khalf + 1];
    // B fragments: 4x16 f32; lane holds W[k+khalf .. +1][ncol]
    v2f bl, bs;
    bl.x = lin_w [(k + khalf)     * HC + ncol];
    bl.y = lin_w [(k + khalf + 1) * HC + ncol];
    bs.x = skip_w[(k + khalf)     * HC + ncol];
    bs.y = skip_w[(k + khalf + 1) * HC + ncol];
    acc_lin  = __builtin_amdgcn_wmma_f32_16x16x4_f32(false, a, false, bl,
                                                     (short)0, acc_lin, false, false);
    acc_skip = __builtin_amdgcn_wmma_f32_16x16x4_f32(false, a, false, bs,
                                                     (short)0, acc_skip, false, false);
  }

  // D layout: VGPR v, lanes 0-15 -> M=v, lanes 16-31 -> M=v+8; N = m16.
  const int rhalf = (lane >> 4) * 8;
  const float sb = skip_b[ncol] + gat_bias[ncol];
  #pragma unroll
  for (int v = 0; v < 8; ++v) {
    const long r = row0 + rhalf + v;
    xh[r * HC + ncol]       = acc_lin[v];
    base_out[r * HC + ncol] = acc_skip[v] + sb;
  }

  // Head-dot reductions: a_src[n, wave] = sum_c xh[n, wave, c] * att_src[wave, c]
  const float ats = att_src[wave * CC + m16];
  const float atd = att_dst[wave * CC + m16];
  v8f ds, dd;
  #pragma unroll
  for (int v = 0; v < 8; ++v) { ds[v] = acc_lin[v] * ats; dd[v] = acc_lin[v] * atd; }
  #pragma unroll
  for (int off = 1; off < 16; off <<= 1) {
    #pragma unroll
    for (int v = 0; v < 8; ++v) {
      ds[v] += __shfl_xor(ds[v], off, 32);
      dd[v] += __shfl_xor(dd[v], off, 32);
    }
  }
  if (m16 == 0) {  // lane 0 writes rows 0..7, lane 16 writes rows 8..15
    #pragma unroll
    for (int v = 0; v < 8; ++v) {
      const long r = row0 + rhalf + v;
      a_src[r * HH + wave] = ds[v];
      a_dst[r * HH + wave] = dd[v];
    }
  }
}

// Order-preserving float->uint encoding for atomic max.
__device__ __forceinline__ unsigned f32_order(float f) {
  unsigned u = __float_as_uint(f);
  return (u & 0x80000000u) ? ~u : (u | 0x80000000u);
}
__device__ __forceinline__ float f32_unorder(unsigned u) {
  return (u & 0x80000000u) ? __uint_as_float(u ^ 0x80000000u)
                           : __uint_as_float(~u);
}

__device__ __forceinline__ float lrelu(float v) {
  return v > 0.0f ? v : NEG_SLOPE * v;
}

// ---------------------------------------------------------------------------
// Kernel 2: segment max of leaky-relu logits over dst. One thread per (edge,h).
// ---------------------------------------------------------------------------
__global__ __launch_bounds__(256)
void gat_segmax_kernel(const long long* __restrict__ ei,
                       const float* __restrict__ a_src,
                       const float* __restrict__ a_dst,
                       unsigned* __restrict__ mord,
                       int E, int total) {
  const long t = (long)blockIdx.x * blockDim.x + threadIdx.x;
  if (t >= (long)total * HH) return;
  const int  h = (int)(t & (HH - 1));
  const long e = t >> 3;
  long src, dst;
  if (e < E) { src = ei[e]; dst = ei[E + e]; } else { src = dst = e - E; }
  const float v = lrelu(a_src[src * HH + h] + a_dst[dst * HH + h]);
  atomicMax(mord + dst * HH + h, f32_order(v));
}

// ---------------------------------------------------------------------------
// Kernel 3: unnormalized weighted scatter. One wave per edge; each lane owns
// 4 of the 128 (h,c) slots. agg[dst] += p * xh[src]; denom[dst,h] += p.
// xh (51 MB) and agg (51 MB) are L2-resident on MI455X (192 MB L2).
// ---------------------------------------------------------------------------
__global__ __launch_bounds__(256)
void gat_scatter_kernel(const long long* __restrict__ ei,
                        const float* __restrict__ a_src,
                        const float* __restrict__ a_dst,
                        const unsigned* __restrict__ mord,
                        const float* __restrict__ xh,
                        float* __restrict__ agg,
                        float* __restrict__ denom,
                        int E, int total) {
  const long wid  = ((long)blockIdx.x * blockDim.x + threadIdx.x) >> 5;
  const int  lane = threadIdx.x & 31;
  if (wid >= total) return;
  long src, dst;
  if (wid < E) { src = ei[wid]; dst = ei[E + wid]; } else { src = dst = wid - E; }

  const float* __restrict__ as = a_src + src * HH;
  const float* __restrict__ ad = a_dst + dst * HH;
  const unsigned* __restrict__ mo = mord + dst * HH;
  const float* __restrict__ xs = xh + src * HC;
  float* __restrict__ ag = agg + dst * HC;

  #pragma unroll
  for (int q = 0; q < 4; ++q) {
    const int hc = lane + 32 * q;
    const int h  = hc >> 4;
    const float v = lrelu(as[h] + ad[h]);
    const float p = __expf(v - f32_unorder(mo[h]));
    atomicAdd(ag + hc, p * xs[hc]);
  }
  if (lane < HH) {
    const float v = lrelu(as[lane] + ad[lane]);
    atomicAdd(denom + dst * HH + lane, __expf(v - f32_unorder(mo[lane])));
  }
}

// ---------------------------------------------------------------------------
// Kernel 4: out = ELU(base + agg/(denom+1e-16)); base already in d_out.
// ---------------------------------------------------------------------------
__global__ __launch_bounds__(256)
void gat_finalize_kernel(const float* __restrict__ agg,
                         const float* __restrict__ denom,
                         float* __restrict__ out, long total) {
  const long t = (long)blockIdx.x * blockDim.x + threadIdx.x;
  if (t >= total) return;
  const int  hc = (int)(t & (HC - 1));
  const long n  = t >> 7;
  const float val = out[t] + agg[t] / (denom[n * HH + (hc >> 4)] + 1e-16f);
  out[t] = val > 0.0f ? val : expm1f(val);
}

extern "C" void kernel_launch(void* const* d_in, const int* in_sizes, int n_in,
                              void* d_out, int out_size, void* d_ws, size_t ws_size,
                              hipStream_t stream) {
  const float*     x        = (const float*)d_in[0];
  const long long* ei       = (const long long*)d_in[1];   // int64 (2, E)
  const float*     lin_w    = (const float*)d_in[2];
  const float*     att_src  = (const float*)d_in[3];
  const float*     att_dst  = (const float*)d_in[4];
  const float*     gat_bias = (const float*)d_in[5];
  const float*     skip_w   = (const float*)d_in[6];
  const float*     skip_b   = (const float*)d_in[7];
  float*           out      = (float*)d_out;

  const int N = in_sizes[0] / INCH;
  const int E = in_sizes[1] / 2;
  const int total = E + N;                   // edges incl. self loops

  // Workspace carve-up
  char* ws = (char*)d_ws;
  float*    xh    = (float*)ws;                 ws += (size_t)N * HC * sizeof(float);
  float*    agg   = (float*)ws;                 ws += (size_t)N * HC * sizeof(float);
  float*    a_src = (float*)ws;                 ws += (size_t)N * HH * sizeof(float);
  float*    a_dst = (float*)ws;                 ws += (size_t)N * HH * sizeof(float);
  unsigned* mord  = (unsigned*)ws;              ws += (size_t)N * HH * sizeof(unsigned);
  float*    denom = (float*)ws;                 ws += (size_t)N * HH * sizeof(float);

  // Re-init accumulators every call (graph-capture safe).
  hipMemsetAsync(agg,   0, (size_t)N * HC * sizeof(float),    stream);
  hipMemsetAsync(mord,  0, (size_t)N * HH * sizeof(unsigned), stream); // 0 == -inf in order encoding
  hipMemsetAsync(denom, 0, (size_t)N * HH * sizeof(float),    stream);

  // 1. Dual WMMA GEMM + head dots (N is a multiple of 16: 100000/16 = 6250)
  gat_gemm_kernel<<<N / 16, 256, 0, stream>>>(
      x, lin_w, skip_w, att_src, att_dst, gat_bias, skip_b,
      xh, a_src, a_dst, out);

  // 2. Segment max
  {
    long items = (long)total * HH;
    int blocks = (int)((items + 255) / 256);
    gat_segmax_kernel<<<blocks, 256, 0, stream>>>(ei, a_src, a_dst, mord, E, total);
  }

  // 3. Weighted scatter (one wave per edge)
  {
    int blocks = (total + 7) / 8;   // 8 waves per 256-thread block
    gat_scatter_kernel<<<blocks, 256, 0, stream>>>(
        ei, a_src, a_dst, mord, xh, agg, denom, E, total);
  }

  // 4. Normalize + bias/skip + ELU
  {
    long items = (long)N * HC;
    int blocks = (int)((items + 255) / 256);
    gat_finalize_kernel<<<blocks, 256, 0, stream>>>(agg, denom, out, items);
  }
}